// DynkMaxGate_29575144800914
// MI455X (gfx1250) — compile-verified
//
#include <hip/hip_runtime.h>
#include <hip/hip_bf16.h>

typedef __attribute__((ext_vector_type(2))) float v2f;
typedef __attribute__((ext_vector_type(8))) float v8f;

#define TOKENS      32768
#define HIDDEN      2048
#define EXPERTS     16
#define TAU         0.5f
#define BLW         0.01f

#define WAVES_PER_BLOCK  8
#define TOKENS_PER_BLOCK (WAVES_PER_BLOCK * 16)       // 128
#define NBLOCKS          (TOKENS / TOKENS_PER_BLOCK)  // 256

// Main kernel: one wave computes a 16-token x 16-expert logits tile with
// V_WMMA_F32_16X16X4_F32, then does softmax + threshold in registers via
// 16-lane shuffle reductions, writes expert_weights, and accumulates
// per-expert partial sums (sum of expert_weights, sum of probs) per block.
__global__ __launch_bounds__(256) void gate_main_kernel(
    const float* __restrict__ x,        // [TOKENS, HIDDEN]
    const float* __restrict__ w,        // [EXPERTS, HIDDEN]
    float* __restrict__ out,            // [TOKENS, EXPERTS]
    float* __restrict__ partial)        // [NBLOCKS, 32]  (16 ew-sums, 16 p-sums)
{
    __shared__ float s_ew[WAVES_PER_BLOCK][EXPERTS];
    __shared__ float s_p [WAVES_PER_BLOCK][EXPERTS];

    const int lane = threadIdx.x & 31;
    const int wave = threadIdx.x >> 5;

    const int tokBase = blockIdx.x * TOKENS_PER_BLOCK + wave * 16;
    const int m    = lane & 15;          // token row within tile / expert col
    const int koff = (lane >> 4) * 2;    // K slot offset: lanes 0-15 -> {k,k+1}, 16-31 -> {k+2,k+3}

    const float* __restrict__ xrow = x + (size_t)(tokBase + m) * HIDDEN + koff;
    const float* __restrict__ wrow = w + (size_t)m * HIDDEN + koff;

    v8f c = {};  // logits accumulator: VGPR r, lanes 0-15 -> token r; lanes 16-31 -> token r+8
#pragma unroll 8
    for (int k = 0; k < HIDDEN; k += 4) {
        v2f a = *(const v2f*)(xrow + k);   // A: 16x4 f32 tile of X
        v2f b = *(const v2f*)(wrow + k);   // B: 4x16 f32 tile of W^T
        c = __builtin_amdgcn_wmma_f32_16x16x4_f32(
                /*neg_a=*/false, a, /*neg_b=*/false, b,
                /*c_mod=*/(short)0, c, /*reuse_a=*/false, /*reuse_b=*/false);
    }

    // Per-row softmax + threshold. Each token's 16 expert logits live in one
    // 16-lane group (lanes 0-15 or 16-31) of a single C VGPR.
    float ew_sum = 0.0f;   // per-(lane&15) expert: sum of expert_weights over my 8 tokens
    float p_sum  = 0.0f;   // per-(lane&15) expert: sum of probs over my 8 tokens
#pragma unroll
    for (int r = 0; r < 8; ++r) {
        float l = c[r];
        // max over the 16-lane group
        float mx = l;
        mx = fmaxf(mx, __shfl_xor(mx, 8, 32));
        mx = fmaxf(mx, __shfl_xor(mx, 4, 32));
        mx = fmaxf(mx, __shfl_xor(mx, 2, 32));
        mx = fmaxf(mx, __shfl_xor(mx, 1, 32));
        float e = __expf(l - mx);          // e at argmax is exactly 1.0
        float s = e;
        s += __shfl_xor(s, 8, 32);
        s += __shfl_xor(s, 4, 32);
        s += __shfl_xor(s, 2, 32);
        s += __shfl_xor(s, 1, 32);
        float p    = e / s;                // prob
        float pmax = 1.0f / s;             // max prob == exp(0)/s
        float thr  = TAU * pmax;
        float ewv  = (p < thr) ? 0.0f : 1.0f;   // straight-through forward value

        const int tok = tokBase + r + ((lane >> 4) << 3);
        out[(size_t)tok * EXPERTS + (lane & 15)] = ewv;
        ew_sum += ewv;
        p_sum  += p;
    }

    // Fold the two token-halves (lane L and L^16 hold the same expert column)
    ew_sum += __shfl_xor(ew_sum, 16, 32);
    p_sum  += __shfl_xor(p_sum,  16, 32);
    if (lane < EXPERTS) {
        s_ew[wave][lane] = ew_sum;
        s_p [wave][lane] = p_sum;
    }
    __syncthreads();

    // Deterministic fixed-order block reduction, one value per expert.
    if (threadIdx.x < EXPERTS) {
        float a = 0.0f, b = 0.0f;
#pragma unroll
        for (int v = 0; v < WAVES_PER_BLOCK; ++v) {
            a += s_ew[v][threadIdx.x];
            b += s_p [v][threadIdx.x];
        }
        partial[(size_t)blockIdx.x * 32 + threadIdx.x]           = a;
        partial[(size_t)blockIdx.x * 32 + EXPERTS + threadIdx.x] = b;
    }
}

// Second kernel: deterministic sequential fold of per-block partials into the
// scalar balance loss.
__global__ __launch_bounds__(32) void gate_loss_kernel(
    const float* __restrict__ partial,     // [NBLOCKS, 32]
    const float* __restrict__ size_rate,   // [EXPERTS]
    float* __restrict__ out_loss)          // scalar at d_out[TOKENS*EXPERTS]
{
    __shared__ float s[32];
    const int t = threadIdx.x;
    float acc = 0.0f;
    for (int b = 0; b < NBLOCKS; ++b)
        acc += partial[(size_t)b * 32 + t];
    s[t] = acc;
    __syncthreads();
    if (t == 0) {
        const float invT = 1.0f / (float)TOKENS;
        float loss = 0.0f;
#pragma unroll
        for (int e = 0; e < EXPERTS; ++e) {
            float token_frac = s[e] * invT * size_rate[e];
            float prob_frac  = s[EXPERTS + e] * invT;
            loss += token_frac * prob_frac;
        }
        loss *= (float)TOKENS * BLW;
        *out_loss = loss;
    }
}

extern "C" void kernel_launch(void* const* d_in, const int* in_sizes, int n_in,
                              void* d_out, int out_size, void* d_ws, size_t ws_size,
                              hipStream_t stream) {
    const float* x         = (const float*)d_in[0];  // routing_inputs [32768,2048]
    const float* w         = (const float*)d_in[1];  // w_logits [16,2048]
    const float* size_rate = (const float*)d_in[2];  // [16]
    float* out     = (float*)d_out;                  // [32768*16] ew + [1] loss
    float* partial = (float*)d_ws;                   // NBLOCKS*32 floats = 32 KB

    gate_main_kernel<<<NBLOCKS, 256, 0, stream>>>(x, w, out, partial);
    gate_loss_kernel<<<1, 32, 0, stream>>>(partial, size_rate,
                                           out + (size_t)TOKENS * EXPERTS);
}